// AutocorrelationToCompositeSinusoidalModelCoefficients_17944373362786
// MI455X (gfx1250) — compile-verified
//
#include <hip/hip_runtime.h>
#include <hip/hip_bf16.h>
#include <math.h>

// ---------------------------------------------------------------------------
// Stage A: u = r @ C  (65536x30) @ (30x30), f32, via V_WMMA_F32_16X16X4_F32.
// One wave per 16-row M-tile; two 16-wide N-tiles (cols 0-15, 16-31; 30 real).
// K = 30 padded to 32, 8 k-steps of 4. All operand loads are address-clamped
// and value-masked (branchless) so EXEC stays all-1s around the WMMAs.
// ---------------------------------------------------------------------------
typedef __attribute__((ext_vector_type(2))) float v2f;
typedef __attribute__((ext_vector_type(8))) float v8f;

__global__ __launch_bounds__(128) void csm_gemm_wmma(
    const float* __restrict__ R, const float* __restrict__ Cm,
    float* __restrict__ U, int batch) {
  const int lane  = threadIdx.x & 31;
  const int wave  = threadIdx.x >> 5;
  const int gw    = blockIdx.x * 4 + wave;          // global wave = M-tile id
  const int l16   = lane & 15;
  const int half  = lane >> 4;                      // 0: K pair 0-1, 1: K pair 2-3
  const int row   = gw * 16 + l16;                  // batch row for A (both halves)
  const int rowc  = (row < batch) ? row : (batch - 1);   // clamped, always valid

  v8f acc0 = {0.f,0.f,0.f,0.f,0.f,0.f,0.f,0.f};
  v8f acc1 = {0.f,0.f,0.f,0.f,0.f,0.f,0.f,0.f};

  const int c1  = 16 + l16;                         // N-tile-1 column
  const int c1c = (c1 < 30) ? c1 : 29;              // clamped column
  const bool c1ok = (c1 < 30);

#pragma unroll
  for (int k = 0; k < 8; ++k) {
    const int kk  = k * 4 + half * 2;               // this lane's K base
    const int k0  = (kk     < 30) ? kk     : 29;    // clamped K indices
    const int k1  = (kk + 1 < 30) ? kk + 1 : 29;
    const bool k0ok = (kk     < 30);
    const bool k1ok = (kk + 1 < 30);

    // Unconditional loads from clamped (in-bounds) addresses, value-masked.
    float a0 = R[(size_t)rowc * 30 + k0];
    float a1 = R[(size_t)rowc * 30 + k1];
    float b00 = Cm[k0 * 30 + l16];
    float b01 = Cm[k1 * 30 + l16];
    float b10 = Cm[k0 * 30 + c1c];
    float b11 = Cm[k1 * 30 + c1c];

    v2f a, b0, b1;
    a.x  = k0ok ? a0 : 0.0f;
    a.y  = k1ok ? a1 : 0.0f;
    b0.x = k0ok ? b00 : 0.0f;
    b0.y = k1ok ? b01 : 0.0f;
    b1.x = (k0ok && c1ok) ? b10 : 0.0f;
    b1.y = (k1ok && c1ok) ? b11 : 0.0f;

    // (neg_a, A, neg_b, B, c_mod, C, reuse_a, reuse_b)
    acc0 = __builtin_amdgcn_wmma_f32_16x16x4_f32(false, a, false, b0,
                                                 (short)0, acc0, false, false);
    acc1 = __builtin_amdgcn_wmma_f32_16x16x4_f32(false, a, false, b1,
                                                 (short)0, acc1, false, false);
  }

  // C/D layout: VGPR v, lanes 0-15 -> row v, lanes 16-31 -> row v+8; col = l16.
#pragma unroll
  for (int v = 0; v < 8; ++v) {
    const int orow = gw * 16 + v + half * 8;
    if (orow < batch) {
      U[(size_t)orow * 30 + l16] = acc0[v];
      if (c1ok) U[(size_t)orow * 30 + c1] = acc1[v];
    }
  }
}

// ---------------------------------------------------------------------------
// Stage B: per batch element (1 thread each, f64):
//   Hankel solve (Prony), polynomial real-root bracketing + bisection,
//   w = acos(x), Vandermonde solve for amplitudes.
// Per-thread 15x16 augmented system in LDS, SoA layout (slot*32 + tid):
// a b64 access from 32 lanes touches all 64 banks exactly once.
// ---------------------------------------------------------------------------
__device__ __forceinline__ void ge15(double* S, int tid) {
  // Augmented 15x16 at slots 0..239; partial-pivot GE; solution -> column 15.
  auto E = [&](int i, int j) -> double& { return S[(i * 16 + j) * 32 + tid]; };
  for (int c = 0; c < 15; ++c) {
    int piv = c;
    double pv = fabs(E(c, c));
    for (int r = c + 1; r < 15; ++r) {
      double v = fabs(E(r, c));
      if (v > pv) { pv = v; piv = r; }
    }
    if (piv != c) {
      for (int j = c; j < 16; ++j) {
        double t = E(c, j); E(c, j) = E(piv, j); E(piv, j) = t;
      }
    }
    const double inv = 1.0 / E(c, c);
    for (int r = c + 1; r < 15; ++r) {
      const double f = E(r, c) * inv;
      for (int j = c + 1; j < 16; ++j) E(r, j) -= f * E(c, j);
    }
  }
  for (int r = 14; r >= 0; --r) {
    double acc = E(r, 15);
    for (int j = r + 1; j < 15; ++j) acc -= E(r, j) * E(j, 15);
    E(r, 15) = acc / E(r, r);
  }
}

__global__ __launch_bounds__(32) void csm_solve(
    const float* __restrict__ Uin, const float* __restrict__ R,
    const float* __restrict__ Cm, float* __restrict__ out,
    int use_ws, int batch) {
  __shared__ double S[255 * 32];                    // 65,280 B
  const int tid = threadIdx.x;
  const int b   = blockIdx.x * 32 + tid;
  if (b >= batch) return;

  auto SL = [&](int slot) -> double& { return S[slot * 32 + tid]; };

  // ---- u[0..29] in f64 ----
  double ud[30];
  if (use_ws) {
#pragma unroll
    for (int i = 0; i < 30; ++i) ud[i] = (double)Uin[(size_t)b * 30 + i];
  } else {
#pragma unroll
    for (int i = 0; i < 30; ++i) {
      double acc = 0.0;
      for (int k = 0; k < 30; ++k)
        acc += (double)R[(size_t)b * 30 + k] * (double)Cm[k * 30 + i];
      ud[i] = acc;
    }
  }

  // ---- Hankel system: (-u[i+j]) p = u[15+i] ----
#pragma unroll
  for (int i = 0; i < 15; ++i) {
#pragma unroll
    for (int j = 0; j < 15; ++j) SL(i * 16 + j) = -ud[i + j];
    SL(i * 16 + 15) = ud[15 + i];
  }
  ge15(S, tid);

  double p[15];
#pragma unroll
  for (int i = 0; i < 15; ++i) p[i] = SL(i * 16 + 15);

  // Companion char-poly: P(x) = x^15 + sum_i p[i] x^i  (no flip needed).
  auto evalP = [&](double x) -> double {
    double v = x + p[14];
#pragma unroll
    for (int i = 13; i >= 0; --i) v = v * x + p[i];
    return v;
  };

  // ---- 15 real roots x=cos(w), w in (0,pi), min separation ~0.095 rad.
  // Scan 128 w-grid points (spacing 0.0245) with a cos/sin rotation
  // recurrence, then 50 bisections per bracket (f64-exact).
  const double PI = 3.14159265358979323846;
  const int NG = 128;
  const double dw = PI / NG;
  const double cd = cos(dw), sd = sin(dw);
  double cw = 1.0, sw = 0.0;
  double xprev = 1.0;
  double fprev = evalP(1.0);
  int count = 0;
  for (int g = 1; g <= NG; ++g) {
    const double cn = cw * cd - sw * sd;
    const double sn = sw * cd + cw * sd;
    cw = cn; sw = sn;
    const double x = cw;
    const double f = evalP(x);
    if (count < 15 && ((f < 0.0) != (fprev < 0.0))) {
      double lo = x, hi = xprev;                    // x decreasing along scan
      const bool flo_neg = (f < 0.0);
      for (int it = 0; it < 50; ++it) {
        const double mid = 0.5 * (lo + hi);
        const double fm = evalP(mid);
        if ((fm < 0.0) == flo_neg) lo = mid; else hi = mid;
      }
      const double root = 0.5 * (lo + hi);
      SL(240 + count) = root;                       // slots 240..254: roots
      out[(size_t)b * 30 + count] = (float)acos(root);  // w ascending
      count++;
    }
    xprev = x; fprev = f;
  }
  for (int i = count; i < 15; ++i) {                // degenerate guard
    SL(240 + i) = -0.95 + 0.013 * i;
    out[(size_t)b * 30 + i] = 0.0f;
  }

  // ---- Vandermonde: V[i][j] = x_j^i,  V m = u[0..14] ----
#pragma unroll
  for (int j = 0; j < 15; ++j) {
    const double xj = SL(240 + j);
    double xx = 1.0;
#pragma unroll
    for (int i = 0; i < 15; ++i) { SL(i * 16 + j) = xx; xx *= xj; }
  }
#pragma unroll
  for (int i = 0; i < 15; ++i) SL(i * 16 + 15) = ud[i];
  ge15(S, tid);
#pragma unroll
  for (int i = 0; i < 15; ++i)
    out[(size_t)b * 30 + 15 + i] = (float)SL(i * 16 + 15);
}

// ---------------------------------------------------------------------------
extern "C" void kernel_launch(void* const* d_in, const int* in_sizes, int n_in,
                              void* d_out, int out_size, void* d_ws, size_t ws_size,
                              hipStream_t stream) {
  const float* R  = (const float*)d_in[0];   // (BATCH, 30)
  const float* Cm = (const float*)d_in[1];   // (30, 30)
  float* out = (float*)d_out;                // (BATCH, 30)
  const int batch = in_sizes[0] / 30;

  const size_t need = (size_t)batch * 30 * sizeof(float);
  const int use_ws = (ws_size >= need) ? 1 : 0;
  float* U = (float*)d_ws;

  if (use_ws) {
    const int waves  = (batch + 15) / 16;    // one wave per 16-row tile
    const int blocks = (waves + 3) / 4;      // 4 waves / 128-thread block
    csm_gemm_wmma<<<blocks, 128, 0, stream>>>(R, Cm, U, batch);
  }
  const int blocksB = (batch + 31) / 32;
  csm_solve<<<blocksB, 32, 0, stream>>>(U, R, Cm, out, use_ws, batch);
}